// GCN_20289425506395
// MI455X (gfx1250) — compile-verified
//
#include <hip/hip_runtime.h>

#define N_NODES  40000
#define N_EDGES  640000
#define N_GRAPHS 64

typedef __attribute__((ext_vector_type(2))) float v2f;
typedef __attribute__((ext_vector_type(8))) float v8f;

__global__ void zero_f32(float* __restrict__ p, int n) {
    int i = blockIdx.x * blockDim.x + threadIdx.x;
    if (i < n) p[i] = 0.0f;
}

// Zero-pad copy: in (K x F) -> out (rows x FP), pads with 0.
// Launch with total = outRows * FP; reads in[] only where r<K && c<F.
__global__ void pad_matrix(const float* __restrict__ in, float* __restrict__ out,
                           int K, int F, int FP, int total) {
    int t = blockIdx.x * blockDim.x + threadIdx.x;
    if (t >= total) return;
    int r = t / FP;
    int c = t - r * FP;
    out[t] = (r < K && c < F) ? in[r * F + c] : 0.0f;
}

// Layer 1 edge aggregation: x is N x 7 (tight), agg is N x 8 (padded, col 7 = 0).
__global__ void edge_aggregate_l1(const float* __restrict__ x,
                                  const int*   __restrict__ src,
                                  const int*   __restrict__ dst,
                                  const float* __restrict__ ew,
                                  float*       __restrict__ agg) {
    int t = blockIdx.x * blockDim.x + threadIdx.x;
    if (t >= N_EDGES * 7) return;
    int e = t / 7;
    int f = t - e * 7;
    float v = x[src[e] * 7 + f] * ew[e];
    atomicAdd(&agg[(dst[e] << 3) + f], v);
}

// Layers 2-5: one thread per (edge, 4-feature group); float4 gather, 4 atomics.
// F is a power of two (8/16/32/64) so rows are 16B-aligned. Node tables are
// L2-resident (<=10MB), so gathers and atomics stay on-chip.
template<int F>
__global__ void edge_aggregate_v4(const float* __restrict__ x,
                                  const int*   __restrict__ src,
                                  const int*   __restrict__ dst,
                                  const float* __restrict__ ew,
                                  float*       __restrict__ agg) {
    constexpr int FQ = F / 4;
    int t = blockIdx.x * blockDim.x + threadIdx.x;
    if (t >= N_EDGES * FQ) return;
    int e = t / FQ;
    int q = t - e * FQ;
    float w = ew[e];
    const float4 v = *(const float4*)(x + src[e] * F + q * 4);
    float* a = agg + dst[e] * F + q * 4;
    atomicAdd(a + 0, v.x * w);
    atomicAdd(a + 1, v.y * w);
    atomicAdd(a + 2, v.z * w);
    atomicAdd(a + 3, v.w * w);
}

// One wave computes one 16x16 output tile of out = agg@w_rel + b_rel + xin@w_root
// with exact-fp32 V_WMMA_F32_16X16X4_F32. All operands padded => no load guards.
//   A (16x4): lane l, vgpr v -> M=l%16, K=2*(l/16)+v
//   B (4x16): lane l, vgpr v -> K=2*(l/16)+v, N=l%16
//   C/D     : lane l, vgpr i -> M=i+8*(l/16), N=l%16
template<int KP, int FOUT, int FOUTP>
__global__ void graphconv_gemm(const float* __restrict__ agg,   // N x KP
                               const float* __restrict__ xin,   // N x KP
                               const float* __restrict__ wrp,   // KP x FOUTP (padded)
                               const float* __restrict__ brp,   // FOUTP (padded)
                               const float* __restrict__ wop,   // KP x FOUTP (padded)
                               float*       __restrict__ out) { // N x FOUT
    constexpr int NT = FOUTP / 16;
    constexpr int MT = N_NODES / 16;      // 2500, exact
    int wave = blockIdx.x * (blockDim.x >> 5) + (threadIdx.x >> 5);
    int lane = threadIdx.x & 31;
    int tm = wave / NT;
    int tn = wave - tm * NT;
    if (tm >= MT) return;                 // wave-uniform

    int khi = (lane >> 4) << 1;           // 0 or 2
    int aOff = (tm * 16 + (lane & 15)) * KP + khi;   // even => v2f aligned
    int cB   = tn * 16 + (lane & 15);

    v8f c = {};

    #pragma unroll
    for (int k0 = 0; k0 < KP; k0 += 4) {
        v2f a = *(const v2f*)(agg + aOff + k0);
        v2f b;
        b[0] = wrp[(khi + k0) * FOUTP + cB];
        b[1] = wrp[(khi + k0 + 1) * FOUTP + cB];
        c = __builtin_amdgcn_wmma_f32_16x16x4_f32(false, a, false, b,
                                                  (short)0, c, false, false);
    }
    #pragma unroll
    for (int k0 = 0; k0 < KP; k0 += 4) {
        v2f a = *(const v2f*)(xin + aOff + k0);
        v2f b;
        b[0] = wop[(khi + k0) * FOUTP + cB];
        b[1] = wop[(khi + k0 + 1) * FOUTP + cB];
        c = __builtin_amdgcn_wmma_f32_16x16x4_f32(false, a, false, b,
                                                  (short)0, c, false, false);
    }

    float bias = brp[cB];
    int rbase = tm * 16 + ((lane >> 4) << 3);
    if (FOUT == FOUTP || cB < FOUT) {     // folds away except layer 1
        #pragma unroll
        for (int i = 0; i < 8; ++i)
            out[(rbase + i) * FOUT + cB] = c[i] + bias;
    }
}

__global__ void pool_sum_v4(const float* __restrict__ h,
                            const int*   __restrict__ batch,
                            float*       __restrict__ pooled) {
    int t = blockIdx.x * blockDim.x + threadIdx.x;
    if (t >= N_NODES * 32) return;        // 128/4 feature groups per node
    int n = t >> 5;
    int q = (t & 31) << 2;
    const float4 v = *(const float4*)(h + (n << 7) + q);
    float* p = pooled + (batch[n] << 7) + q;
    atomicAdd(p + 0, v.x);
    atomicAdd(p + 1, v.y);
    atomicAdd(p + 2, v.z);
    atomicAdd(p + 3, v.w);
}

__global__ void final_linear(const float* __restrict__ pooled,
                             const float* __restrict__ w_lin,
                             const float* __restrict__ b_lin,
                             float*       __restrict__ out) {
    int t = threadIdx.x;            // 128 threads: 64 graphs x 2 classes
    int g = t >> 1;
    int c = t & 1;
    float s = b_lin[c];
    #pragma unroll 8
    for (int f = 0; f < 128; ++f)
        s += pooled[(g << 7) + f] * w_lin[2 * f + c];
    out[2 * g + c] = s;
}

extern "C" void kernel_launch(void* const* d_in, const int* in_sizes, int n_in,
                              void* d_out, int out_size, void* d_ws, size_t ws_size,
                              hipStream_t stream) {
    const float* x     = (const float*)d_in[0];
    const int*   ei    = (const int*)  d_in[1];
    const float* ew    = (const float*)d_in[2];
    const int*   batch = (const int*)  d_in[3];
    const float* wr[5], *br[5], *wo[5];
    for (int i = 0; i < 5; ++i) {
        wr[i] = (const float*)d_in[4 + 3 * i];
        br[i] = (const float*)d_in[5 + 3 * i];
        wo[i] = (const float*)d_in[6 + 3 * i];
    }
    const float* w_lin = (const float*)d_in[19];
    const float* b_lin = (const float*)d_in[20];
    const int* src = ei;
    const int* dst = ei + N_EDGES;

    // ---- workspace layout (floats) ----
    float* bufA    = (float*)d_ws;                          // N x 128
    float* bufB    = bufA + (size_t)N_NODES * 128;          // N x 128
    float* aggb    = bufB + (size_t)N_NODES * 128;          // N x 64 (max Fin)
    float* xpad    = aggb + (size_t)N_NODES * 64;           // N x 8 (layer-1 x padded)
    float* pooled  = xpad + (size_t)N_NODES * 8;            // 64 x 128
    float* scratch = pooled + (size_t)N_GRAPHS * 128;       // padded weights/biases

    // padded params: per layer  wrel(KP x FOUTP) + wroot(KP x FOUTP) + bias(FOUTP)
    const int KINv[5]   = {7, 8, 16, 32, 64};    // true rows in source weights
    const int KPv[5]    = {8, 8, 16, 32, 64};    // padded K (multiple of 4)
    const int FOUTv[5]  = {8, 16, 32, 64, 128};  // true cols
    const int FOUTPv[5] = {16, 16, 32, 64, 128}; // padded cols (multiple of 16)
    float* wrp[5]; float* wop[5]; float* brp[5];
    {
        float* p = scratch;
        for (int i = 0; i < 5; ++i) {
            wrp[i] = p; p += KPv[i] * FOUTPv[i];
            wop[i] = p; p += KPv[i] * FOUTPv[i];
            brp[i] = p; p += FOUTPv[i];
        }
    }

    auto zero = [&](float* p, int n) {
        zero_f32<<<(n + 255) / 256, 256, 0, stream>>>(p, n);
    };

    // ---- prep: zero-pad params (true row/col counts guard all reads) ----
    for (int i = 0; i < 5; ++i) {
        int total = KPv[i] * FOUTPv[i];
        pad_matrix<<<(total + 255) / 256, 256, 0, stream>>>(
            wr[i], wrp[i], KINv[i], FOUTv[i], FOUTPv[i], total);
        pad_matrix<<<(total + 255) / 256, 256, 0, stream>>>(
            wo[i], wop[i], KINv[i], FOUTv[i], FOUTPv[i], total);
        pad_matrix<<<(FOUTPv[i] + 255) / 256, 256, 0, stream>>>(
            br[i], brp[i], 1, FOUTv[i], FOUTPv[i], FOUTPv[i]);
    }
    // layer-1 x: N x 7 -> N x 8 (col 7 = 0)
    pad_matrix<<<(N_NODES * 8 + 255) / 256, 256, 0, stream>>>(
        x, xpad, N_NODES, 7, 8, N_NODES * 8);

    const int MT = N_NODES / 16;   // 2500

    // ---- Layer 1: 7 -> 8  (padded K=8) ----
    zero(aggb, N_NODES * 8);
    edge_aggregate_l1<<<(N_EDGES * 7 + 255) / 256, 256, 0, stream>>>(x, src, dst, ew, aggb);
    graphconv_gemm<8, 8, 16><<<(MT * 1 + 3) / 4, 128, 0, stream>>>(
        aggb, xpad, wrp[0], brp[0], wop[0], bufA);

    // ---- Layer 2: 8 -> 16 ----
    zero(aggb, N_NODES * 8);
    edge_aggregate_v4<8><<<(N_EDGES * 2 + 255) / 256, 256, 0, stream>>>(bufA, src, dst, ew, aggb);
    graphconv_gemm<8, 16, 16><<<(MT * 1 + 3) / 4, 128, 0, stream>>>(
        aggb, bufA, wrp[1], brp[1], wop[1], bufB);

    // ---- Layer 3: 16 -> 32 ----
    zero(aggb, N_NODES * 16);
    edge_aggregate_v4<16><<<(N_EDGES * 4 + 255) / 256, 256, 0, stream>>>(bufB, src, dst, ew, aggb);
    graphconv_gemm<16, 32, 32><<<(MT * 2 + 3) / 4, 128, 0, stream>>>(
        aggb, bufB, wrp[2], brp[2], wop[2], bufA);

    // ---- Layer 4: 32 -> 64 ----
    zero(aggb, N_NODES * 32);
    edge_aggregate_v4<32><<<(N_EDGES * 8 + 255) / 256, 256, 0, stream>>>(bufA, src, dst, ew, aggb);
    graphconv_gemm<32, 64, 64><<<(MT * 4 + 3) / 4, 128, 0, stream>>>(
        aggb, bufA, wrp[3], brp[3], wop[3], bufB);

    // ---- Layer 5: 64 -> 128 ----
    zero(aggb, N_NODES * 64);
    edge_aggregate_v4<64><<<(N_EDGES * 16 + 255) / 256, 256, 0, stream>>>(bufB, src, dst, ew, aggb);
    graphconv_gemm<64, 128, 128><<<(MT * 8 + 3) / 4, 128, 0, stream>>>(
        aggb, bufB, wrp[4], brp[4], wop[4], bufA);

    // ---- Pool + head ----
    zero(pooled, N_GRAPHS * 128);
    pool_sum_v4<<<(N_NODES * 32 + 255) / 256, 256, 0, stream>>>(bufA, batch, pooled);
    final_linear<<<1, 128, 0, stream>>>(pooled, w_lin, b_lin, (float*)d_out);
}